// LRBP_23124103922239
// MI455X (gfx1250) — compile-verified
//
#include <hip/hip_runtime.h>
#include <math.h>

typedef __attribute__((ext_vector_type(2))) float v2f;
typedef __attribute__((ext_vector_type(8))) float v8f;

#define DDIM 512
#define BATCH 256
#define CNUM 10
#define KTOT (DDIM*DDIM)        /* 262144 */
#define NSLICE 128
#define KSLICE (KTOT/NSLICE)    /* 2048 */
#define NT 16                   /* class dim padded to one WMMA tile */

// ---------------- Phase 0: Wc[c, i*D+j] = u1 u1^T - u2 u2^T ----------------
__global__ void __launch_bounds__(256) wc_kernel(const float* __restrict__ w,
                                                 float* __restrict__ wc) {
    int idx = blockIdx.x * blockDim.x + threadIdx.x;
    if (idx >= CNUM * DDIM * DDIM) return;
    int c   = idx / (DDIM * DDIM);
    int rem = idx % (DDIM * DDIM);
    int i   = rem / DDIM;
    int j   = rem % DDIM;
    const float* wi = w + ((size_t)c * DDIM + i) * 6;
    const float* wj = w + ((size_t)c * DDIM + j) * 6;
    float v = wi[0]*wj[0] + wi[1]*wj[1] + wi[2]*wj[2]
            - wi[3]*wj[3] - wi[4]*wj[4] - wi[5]*wj[5];
    wc[idx] = v;
}

// ---------------- Phase A: logits partials via V_WMMA_F32_16X16X4_F32 ------
// wave gw: ks = gw>>4 (K-slice), bt = gw&15 (batch tile of 16 rows)
__global__ void __launch_bounds__(256) gemm_kernel(const float* __restrict__ x,
                                                   const float* __restrict__ wc,
                                                   float* __restrict__ partial) {
    int lane = threadIdx.x & 31;
    int wid  = threadIdx.x >> 5;
    int gw   = blockIdx.x * 8 + wid;     // 0..2047
    int ks   = gw >> 4;                  // 0..127
    int bt   = gw & 15;                  // 0..15
    int half = lane >> 4;                // 0 or 1  (K sub-pair)
    int l15  = lane & 15;
    int row  = bt * 16 + l15;            // batch row (A matrix M index)
    int n    = (l15 < CNUM) ? l15 : 0;   // clamp: cols 10..15 unused downstream

    size_t kbase = (size_t)ks * KSLICE + (size_t)(half * 2);
    const float* xp = x  + (size_t)row * KTOT + kbase;   // A: x[row][k+koff..+1]
    const float* bp = wc + (size_t)n   * KTOT + kbase;   // B: Wc[n][k+koff..+1]

    v8f acc = {0.f,0.f,0.f,0.f,0.f,0.f,0.f,0.f};
    #pragma unroll 8
    for (int k = 0; k < KSLICE; k += 4) {
        v2f a = __builtin_nontemporal_load((const v2f*)(xp + k)); // stream-once
        v2f b = *(const v2f*)(bp + k);                            // L2-resident
        acc = __builtin_amdgcn_wmma_f32_16x16x4_f32(
                  /*neg_a=*/false, a, /*neg_b=*/false, b,
                  /*c_mod=*/(short)0, acc, /*reuse_a=*/false, /*reuse_b=*/false);
    }

    // D layout: VGPR r, lane -> D[M = r + 8*half][N = l15]
    float* outp = partial + ((size_t)ks * BATCH + (size_t)bt * 16) * NT;
    #pragma unroll
    for (int r = 0; r < 8; ++r) {
        int m = r + 8 * half;
        outp[(size_t)m * NT + l15] = acc[r];
    }
}

// ---------------- Phase B: reduce K-slices + bias + softmax ----------------
__global__ void __launch_bounds__(32) softmax_kernel(const float* __restrict__ partial,
                                                     const float* __restrict__ bias,
                                                     float* __restrict__ out) {
    int b    = blockIdx.x;      // 0..255
    int lane = threadIdx.x;     // 0..31
    float s = 0.f;
    if (lane < NT) {
        for (int ksl = 0; ksl < NSLICE; ++ksl)
            s += partial[((size_t)ksl * BATCH + b) * NT + lane];
    }
    bool valid = (lane < CNUM);
    float logit = valid ? (s + bias[lane]) : -INFINITY;

    float m = logit;
    for (int off = 16; off > 0; off >>= 1)
        m = fmaxf(m, __shfl_xor(m, off, 32));
    float e = valid ? expf(logit - m) : 0.f;
    float tot = e;
    for (int off = 16; off > 0; off >>= 1)
        tot += __shfl_xor(tot, off, 32);
    if (valid) out[(size_t)b * CNUM + lane] = e / tot;
}

extern "C" void kernel_launch(void* const* d_in, const int* in_sizes, int n_in,
                              void* d_out, int out_size, void* d_ws, size_t ws_size,
                              hipStream_t stream) {
    const float* x    = (const float*)d_in[0];   // (256, 512, 512) f32
    const float* w    = (const float*)d_in[1];   // (10, 512, 6)    f32
    const float* bias = (const float*)d_in[2];   // (10,)           f32
    float* out = (float*)d_out;                  // (256, 10)       f32

    float* wc      = (float*)d_ws;                         // 10*262144 f32 = 10.5 MB
    float* partial = wc + (size_t)CNUM * KTOT;             // 128*256*16 f32 = 2 MB

    wc_kernel<<<(CNUM * DDIM * DDIM + 255) / 256, 256, 0, stream>>>(w, wc);
    gemm_kernel<<<256, 256, 0, stream>>>(x, wc, partial);  // 2048 waves
    softmax_kernel<<<BATCH, 32, 0, stream>>>(partial, bias, out);
}